// pathLoss_69441031241965
// MI455X (gfx1250) — compile-verified
//
#include <hip/hip_runtime.h>
#include <math.h>

// Problem constants (from setup_inputs: B=4096, T=2048, C=5).
#define B_ROWS 4096
#define T_STEPS 2048
#define NCLS 5
#define CH_T 128                      // timesteps per staged chunk
#define NCHUNK (T_STEPS / CH_T)       // 16 chunks per row
#define CH_FLOATS (CH_T * NCLS)       // 640 floats = 2560 bytes per chunk
#define WAVES_PER_BLOCK 8
#define BLOCK_THREADS (WAVES_PER_BLOCK * 32)

#if defined(__has_builtin)
#if __has_builtin(__builtin_amdgcn_global_load_async_to_lds_b128)
#define HAVE_ASYNC_LDS 1
#endif
#endif

typedef int v4i __attribute__((vector_size(16)));
typedef __attribute__((address_space(1))) v4i g_v4i;
typedef __attribute__((address_space(3))) v4i l_v4i;

// Generic->AS casts via the integer detour (generic LDS addr low 32 bits are
// the LDS byte offset per the CDNA5 aperture model; global AS1 == generic bits).
__device__ __forceinline__ g_v4i* to_global_v4(const void* p) {
  return (g_v4i*)(unsigned long long)p;
}
__device__ __forceinline__ l_v4i* to_lds_v4(void* p) {
  return (l_v4i*)(unsigned int)(unsigned long long)p;
}

// Stage one 2560-byte chunk (128 timesteps x 5 classes) into LDS.
// 5 pieces of 512B; each of the 32 lanes moves 16B per piece.
__device__ __forceinline__ void issue_chunk(const float* __restrict__ gsrc,
                                            float* lbuf, int lane) {
#ifdef HAVE_ASYNC_LDS
#pragma unroll
  for (int p = 0; p < 5; ++p) {
    const float* g = gsrc + p * 128 + lane * 4;
    float* l = lbuf + p * 128 + lane * 4;
    __builtin_amdgcn_global_load_async_to_lds_b128(to_global_v4(g), to_lds_v4(l),
                                                   /*offset=*/0, /*cpol=*/0);
  }
#else
#pragma unroll
  for (int p = 0; p < 5; ++p) {
    float4 v = *(const float4*)(gsrc + p * 128 + lane * 4);
    *(float4*)(lbuf + p * 128 + lane * 4) = v;
  }
#endif
}

__global__ __launch_bounds__(BLOCK_THREADS) void pathloss_rows_kernel(
    const float* __restrict__ outputs, const int* __restrict__ labels,
    float* __restrict__ row_sums) {
  // cos(0.15 * k) for k in [-2048, 2048]: kills all cosf in the hot loop.
  __shared__ float cosTab[2 * T_STEPS + 1];
  __shared__ __align__(16) float stage[WAVES_PER_BLOCK][2][CH_FLOATS];

  const int lane = threadIdx.x & 31;
  const int wave = threadIdx.x >> 5;
  const int row = blockIdx.x * WAVES_PER_BLOCK + wave;

  for (int i = threadIdx.x; i < 2 * T_STEPS + 1; i += BLOCK_THREADS)
    cosTab[i] = cosf(0.15f * (float)(i - T_STEPS));
  __syncthreads();

  const float* rowOut = outputs + (size_t)row * (T_STEPS * NCLS);
  const int* rowLab = labels + (size_t)row * T_STEPS;

  issue_chunk(rowOut, &stage[wave][0][0], lane);

  int carrySp = 0, carrySt = 0;   // running integer theta-step counts
  float cxp = 0.f, cxt = 0.f;     // running px prefixes
  float acc = 0.f;                // sum |px_p - px_t|

  for (int c = 0; c < NCHUNK; ++c) {
    const float* cur = &stage[wave][c & 1][0];
    if (c + 1 < NCHUNK) {
      issue_chunk(rowOut + (size_t)(c + 1) * CH_FLOATS,
                  &stage[wave][(c + 1) & 1][0], lane);
#ifdef HAVE_ASYNC_LDS
      __builtin_amdgcn_s_wait_asynccnt(5);  // chunk c fully landed in LDS
#endif
    } else {
#ifdef HAVE_ASYNC_LDS
      __builtin_amdgcn_s_wait_asynccnt(0);
#endif
    }
    asm volatile("" ::: "memory");  // keep LDS reads below the wait

#pragma unroll
    for (int tile = 0; tile < CH_T / 32; ++tile) {
      const int tl = tile * 32 + lane;       // t within chunk
      const int t = c * CH_T + tl;           // global t
      const float* e = cur + tl * NCLS;

      // argmax over 5 classes, first-max tiebreak (matches jnp.argmax)
      int pred = 0;
      float best = e[0], v;
      v = e[1]; if (v > best) { best = v; pred = 1; }
      v = e[2]; if (v > best) { best = v; pred = 2; }
      v = e[3]; if (v > best) { best = v; pred = 3; }
      v = e[4]; if (v > best) { best = v; pred = 4; }

      const int lt = rowLab[t];

      // class -> theta step {0,+1,-1,+1,-1} and "moves" {1,0,0,1,1}
      const int sp = (pred == 0) ? 0 : ((pred & 1) ? 1 : -1);
      const int mp = (pred == 0) | (pred >= 3);
      const int st = (lt == 0) ? 0 : ((lt & 1) ? 1 : -1);
      const int mt = (lt == 0) | (lt >= 3);

      // Packed dual integer scan: pred steps in [15:0], label steps in [31:16],
      // each biased by +64 so 32-element partial sums stay in 16 bits.
      int vpk = (sp + 64) | ((st + 64) << 16);
#pragma unroll
      for (int d = 1; d < 32; d <<= 1) {
        int n = __shfl_up(vpk, d, 32);
        if (lane >= d) vpk += n;
      }
      const int bias = 64 * (lane + 1);
      const int Sp = carrySp + ((vpk & 0xFFFF) - bias);
      const int St = carrySt + ((vpk >> 16) - bias);
      const int tot = __shfl(vpk, 31, 32);
      carrySp += (tot & 0xFFFF) - 64 * 32;
      carrySt += (tot >> 16) - 64 * 32;

      float dxp = mp ? cosTab[Sp + T_STEPS] : 0.0f;
      float dxt = mt ? cosTab[St + T_STEPS] : 0.0f;

      // Dual float inclusive scan -> px prefixes within tile.
#pragma unroll
      for (int d = 1; d < 32; d <<= 1) {
        float np = __shfl_up(dxp, d, 32);
        float nt = __shfl_up(dxt, d, 32);
        if (lane >= d) { dxp += np; dxt += nt; }
      }
      const float pxp = cxp + dxp;
      const float pxt = cxt + dxt;
      cxp += __shfl(dxp, 31, 32);
      cxt += __shfl(dxt, 31, 32);

      acc += fabsf(pxp - pxt);
    }
  }

  // Wave reduction of the per-lane partial sums.
#pragma unroll
  for (int d = 16; d > 0; d >>= 1) acc += __shfl_xor(acc, d, 32);
  if (lane == 0) row_sums[row] = acc;
}

__global__ void pathloss_final_reduce(const float* __restrict__ row_sums,
                                      float* __restrict__ out) {
  __shared__ float sm[256];
  float s = 0.f;
  for (int i = threadIdx.x; i < B_ROWS; i += 256) s += row_sums[i];
  sm[threadIdx.x] = s;
  __syncthreads();
  for (int d = 128; d > 0; d >>= 1) {
    if ((int)threadIdx.x < d) sm[threadIdx.x] += sm[threadIdx.x + d];
    __syncthreads();
  }
  if (threadIdx.x == 0) {
    // loss = sqrt(2) * sum|diff| / (B * (T+1)); diff at t=0 is 0.
    out[0] = sm[0] * (sqrtf(2.0f) / ((float)B_ROWS * (float)(T_STEPS + 1)));
  }
}

extern "C" void kernel_launch(void* const* d_in, const int* in_sizes, int n_in,
                              void* d_out, int out_size, void* d_ws, size_t ws_size,
                              hipStream_t stream) {
  (void)in_sizes; (void)n_in; (void)out_size; (void)ws_size;
  const float* outputs = (const float*)d_in[0];
  const int* labels = (const int*)d_in[1];  // integer input -> const int*
  float* row_sums = (float*)d_ws;           // 4096 floats of scratch

  pathloss_rows_kernel<<<B_ROWS / WAVES_PER_BLOCK, BLOCK_THREADS, 0, stream>>>(
      outputs, labels, row_sums);
  pathloss_final_reduce<<<1, 256, 0, stream>>>(row_sums, (float*)d_out);
}